// CUDAKernel_46497315947114
// MI455X (gfx1250) — compile-verified
//
#include <hip/hip_runtime.h>
#include <stdint.h>

typedef __attribute__((ext_vector_type(2))) float v2f;
typedef __attribute__((ext_vector_type(8))) float v8f;
typedef __attribute__((ext_vector_type(4))) int   v4i;

#define U_    128
#define C_    300
#define S_    32
#define NSEG  16
#define P1_   32
#define P2_   64
#define P3_   96
#define PTOT  (P1_ + P2_ + P3_)   /* 192 */
#define NSTEP (PTOT / 4)          /* 48  */
#define STEP1 (P1_ / 4)           /* 8   */
#define STEP12 ((P1_ + P2_) / 4)  /* 24  */
#define UB4   (U_ * 4)            /* 512: bytes per segment/coupling row */

// ---- CDNA5 async global->LDS copy (ASYNCcnt-tracked) --------------------
__device__ __forceinline__ void async_b128(uint32_t lds_off, const void* gptr) {
  asm volatile("global_load_async_to_lds_b128 %0, %1, off"
               :: "v"(lds_off), "v"(gptr) : "memory");
}
__device__ __forceinline__ void wait_async0() {
  asm volatile("s_wait_asynccnt 0x0" ::: "memory");
}

// pair load {p[off], p[off+64B]}: two tiles' channels of one factor.
// Same base + imm-expressible deltas -> ds_load_2addr_b32 / global offset imm.
__device__ __forceinline__ v2f pair_ld(const char* base, int byteoff) {
  v2f r;
  r.x = *(const float*)(base + byteoff);
  r.y = *(const float*)(base + byteoff + 64);
  return r;
}

__global__ __launch_bounds__(128)
void symcon_wmma_kernel(const float* __restrict__ x0,  const int* __restrict__ i0,
                        const float* __restrict__ x1,
                        const int* __restrict__ idx1, const int* __restrict__ w1,
                        const int* __restrict__ o1,   const float* __restrict__ c1,
                        const int* __restrict__ idx2, const int* __restrict__ w2,
                        const int* __restrict__ o2,   const float* __restrict__ c2,
                        const int* __restrict__ idx3, const int* __restrict__ w3,
                        const int* __restrict__ o3,   const float* __restrict__ c3,
                        float* __restrict__ out, int Btot)
{
  __shared__ __align__(16) float x1s[2][S_ * U_];  // 32 KB: two nodes' x1 tiles
  __shared__ v2f  atab[NSTEP * 32];                // 12 KB: A-matrix chunks (step, lane)
  __shared__ v4i  meta[PTOT];                      //  3 KB: byte-scaled {w, idx0, idx1, idx2}

  const int tid = threadIdx.x;
  const int b0  = blockIdx.x * 2;

  // ---- stage both nodes' x1 tiles into LDS with async copies ----
  {
    uint32_t base = (uint32_t)(uintptr_t)&x1s[0][0];
#pragma unroll
    for (int j = 0; j < (2 * S_ * U_ / 4) / 128; ++j) {   // 16 x b128 per thread
      int ch = j * 128 + tid;            // 0..2047
      int nd = ch >> 10;                 // node 0/1
      int c2 = ch & 1023;                // chunk within node
      if (b0 + nd < Btot)
        async_b128(base + ch * 16, x1 + (size_t)(b0 + nd) * (S_ * U_) + c2 * 4);
    }
  }

  // ---- pack per-path metadata (pre-scaled to byte offsets) ----
  for (int p = tid; p < PTOT; p += 128) {
    int w, a0 = 0, a1 = 0, a2 = 0;
    if (p < P1_) {
      w = w1[p]; a0 = idx1[p];
    } else if (p < P1_ + P2_) {
      int q = p - P1_;
      w = w2[q]; a0 = idx2[q * 2]; a1 = idx2[q * 2 + 1];
    } else {
      int q = p - P1_ - P2_;
      w = w3[q]; a0 = idx3[q * 3]; a1 = idx3[q * 3 + 1]; a2 = idx3[q * 3 + 2];
    }
    meta[p] = (v4i){w * UB4, a0 * UB4, a1 * UB4, a2 * UB4};
  }

  // ---- build WMMA A-matrix chunks: A[m, k] = c_p * (o_p == m) ----
  // lanes 0-15 hold K = {4s, 4s+1}; lanes 16-31 hold K = {4s+2, 4s+3}
  for (int e = tid; e < NSTEP * 32; e += 128) {
    int s    = e >> 5;
    int lane = e & 31;
    int m    = lane & 15;
    int hf   = lane >> 4;
    int p0   = 4 * s + 2 * hf;
    float A[2];
#pragma unroll
    for (int t = 0; t < 2; ++t) {
      int p = p0 + t;
      int o; float c;
      if (p < P1_)            { o = o1[p];              c = c1[p]; }
      else if (p < P1_ + P2_) { o = o2[p - P1_];        c = c2[p - P1_]; }
      else                    { o = o3[p - P1_ - P2_];  c = c3[p - P1_ - P2_]; }
      A[t] = (o == m) ? c : 0.0f;
    }
    atab[e] = (v2f){A[0], A[1]};
  }

  wait_async0();
  __syncthreads();

  const int lane = tid & 31;
  const int wv   = tid >> 5;       // 0..3
  const int nd   = wv >> 1;        // node within block (0/1)
  const int wp   = wv & 1;         // channel half (0: u<64, 1: u>=64)
  const int m    = lane & 15;      // row / column index inside 16x16
  const int half = lane >> 4;      // 0: paths 4s,4s+1  1: paths 4s+2,4s+3
  const int bn   = b0 + nd;        // this wave's node
  if (bn >= Btot) return;          // wave-uniform guard (EXEC stays all-ones)

  const int u    = wp * 64 + m;    // tile-A channel; tiles B/C/D at +16/+32/+48
  const int u4   = u * 4;          // byte offset (tiles at +64/+128/+192 bytes)

  const char* g0b = (const char*)x0 + (size_t)i0[bn] * (C_ * U_ * sizeof(float));
  const char* x1b = (const char*)&x1s[nd][0];

  v8f accA = {}, accB = {}, accC = {}, accD = {};

#define DO_STEP(sidx, DEG)                                                    \
  {                                                                           \
    int pa = 4 * (sidx) + 2 * half;                                           \
    v4i ma = meta[pa];                                                        \
    v4i mb = meta[pa + 1];                                                    \
    v2f Am = atab[(sidx) * 32 + lane];                                        \
    v2f taAB = pair_ld(g0b, ma.x + u4)       * pair_ld(x1b, ma.y + u4);       \
    v2f taCD = pair_ld(g0b, ma.x + u4 + 128) * pair_ld(x1b, ma.y + u4 + 128); \
    v2f tbAB = pair_ld(g0b, mb.x + u4)       * pair_ld(x1b, mb.y + u4);       \
    v2f tbCD = pair_ld(g0b, mb.x + u4 + 128) * pair_ld(x1b, mb.y + u4 + 128); \
    if (DEG >= 2) {                                                           \
      taAB *= pair_ld(x1b, ma.z + u4);  taCD *= pair_ld(x1b, ma.z + u4 + 128);\
      tbAB *= pair_ld(x1b, mb.z + u4);  tbCD *= pair_ld(x1b, mb.z + u4 + 128);\
    }                                                                         \
    if (DEG >= 3) {                                                           \
      taAB *= pair_ld(x1b, ma.w + u4);  taCD *= pair_ld(x1b, ma.w + u4 + 128);\
      tbAB *= pair_ld(x1b, mb.w + u4);  tbCD *= pair_ld(x1b, mb.w + u4 + 128);\
    }                                                                         \
    v2f BA = (v2f){taAB.x, tbAB.x};  /* columns u       */                    \
    v2f BB = (v2f){taAB.y, tbAB.y};  /* columns u + 16  */                    \
    v2f BC = (v2f){taCD.x, tbCD.x};  /* columns u + 32  */                    \
    v2f BD = (v2f){taCD.y, tbCD.y};  /* columns u + 48  */                    \
    accA = __builtin_amdgcn_wmma_f32_16x16x4_f32(false, Am, false, BA,        \
                                                 (short)0, accA, false, false);  \
    accB = __builtin_amdgcn_wmma_f32_16x16x4_f32(false, Am, false, BB,        \
                                                 (short)0, accB, false, false);  \
    accC = __builtin_amdgcn_wmma_f32_16x16x4_f32(false, Am, false, BC,        \
                                                 (short)0, accC, false, false);  \
    accD = __builtin_amdgcn_wmma_f32_16x16x4_f32(false, Am, false, BD,        \
                                                 (short)0, accD, false, false);  \
  }

  for (int s = 0;      s < STEP1;  ++s) DO_STEP(s, 1);
  for (int s = STEP1;  s < STEP12; ++s) DO_STEP(s, 2);
  for (int s = STEP12; s < NSTEP;  ++s) DO_STEP(s, 3);
#undef DO_STEP

  // ---- store: lane holds segments half*8 + r for columns u/+16/+32/+48 ----
  float* op = out + (size_t)bn * (NSEG * U_) + (size_t)(half * 8) * U_;
#pragma unroll
  for (int r = 0; r < 8; ++r) {
    op[r * U_ + u]      = accA[r];
    op[r * U_ + u + 16] = accB[r];
    op[r * U_ + u + 32] = accC[r];
    op[r * U_ + u + 48] = accD[r];
  }
}

extern "C" void kernel_launch(void* const* d_in, const int* in_sizes, int n_in,
                              void* d_out, int out_size, void* d_ws, size_t ws_size,
                              hipStream_t stream) {
  const float* x0   = (const float*)d_in[0];
  const int*   i0   = (const int*)  d_in[1];
  const float* x1   = (const float*)d_in[2];
  const int*   idx1 = (const int*)  d_in[3];
  const int*   w1   = (const int*)  d_in[4];
  const int*   o1   = (const int*)  d_in[5];
  const float* c1   = (const float*)d_in[6];
  const int*   idx2 = (const int*)  d_in[7];
  const int*   w2   = (const int*)  d_in[8];
  const int*   o2   = (const int*)  d_in[9];
  const float* c2   = (const float*)d_in[10];
  const int*   idx3 = (const int*)  d_in[11];
  const int*   w3   = (const int*)  d_in[12];
  const int*   o3   = (const int*)  d_in[13];
  const float* c3   = (const float*)d_in[14];
  float*       out  = (float*)d_out;

  const int B = in_sizes[1];            // number of nodes (i0 length)
  const int nblk = (B + 1) / 2;         // two nodes per block

  symcon_wmma_kernel<<<nblk, 128, 0, stream>>>(x0, i0, x1,
                                               idx1, w1, o1, c1,
                                               idx2, w2, o2, c2,
                                               idx3, w3, o3, c3,
                                               out, B);
  (void)d_ws; (void)ws_size; (void)n_in; (void)out_size;
}